// ViT_29076928594212
// MI455X (gfx1250) — compile-verified
//
#include <hip/hip_runtime.h>
#include <hip/hip_bf16.h>

typedef float v2f __attribute__((ext_vector_type(2)));
typedef float v8f __attribute__((ext_vector_type(8)));

#define IMG_C 3
#define IMG_H 512
#define IMG_W 512
#define PSZ   8
#define DIN   192          // 3*8*8
#define DOUT  16
#define HW    (IMG_H * IMG_W)          // 262144
#define PATCH_PER_ROW   (IMG_W / PSZ)  // 64
#define PATCHES_PER_IMG (PATCH_PER_ROW * PATCH_PER_ROW) // 4096
#define TILES_PER_PROW  (PATCH_PER_ROW / 16)            // 4
#define TILES_PER_IMG   (PATCHES_PER_IMG / 16)          // 256
#define WAVES_PER_BLOCK 8

// One wave computes a 16(patch) x 16(feature) output tile:
//   D = A(16x192 patch rows) * B(192x16 = w^T) + bias
// as 48 chained V_WMMA_F32_16X16X4_F32 steps.
__global__ __launch_bounds__(256) void vit_patch_embed_wmma(
    const float* __restrict__ img,   // [64,3,512,512]
    const float* __restrict__ w,     // [16,192]
    const float* __restrict__ bias,  // [16]
    float* __restrict__ out)         // [64*4096,16]
{
  const int lane = threadIdx.x & 31;
  const int wave = threadIdx.x >> 5;
  const int tile = blockIdx.x * WAVES_PER_BLOCK + wave;

  const int n   = tile >> 8;            // image index (tile / 256)
  const int rem = tile & 255;
  const int pr  = rem >> 2;             // patch row 0..63
  const int pc0 = (rem & 3) << 4;       // patch col base (0,16,32,48)

  const int nlo = lane & 15;            // M for A-lanes / N for B,C,D-lanes
  const int khi = lane >> 4;            // which K-pair within the K-group of 4

  // A: lane loads patch (pc0+nlo)'s elements d = 4s + 2*khi + {0,1}.
  // d = c*64 + py*8 + px  ->  address inside one image row, contiguous float2.
  const float* abase = img + (size_t)n * (IMG_C * HW)
                           + (size_t)(pr * PSZ) * IMG_W
                           + (pc0 + nlo) * PSZ + khi * 2;
  // B: lane loads w[nlo][4s + 2*khi + {0,1}]  (B[k][n] = w[n][k]).
  const float* bbase = w + nlo * DIN + khi * 2;

  const float bv = bias[nlo];           // C/D column N = nlo
  v8f acc = { bv, bv, bv, bv, bv, bv, bv, bv };

#pragma unroll
  for (int s = 0; s < 48; ++s) {
    const int c   = s >> 4;             // channel     (d0/64)
    const int py  = (s >> 1) & 7;       // row in patch((d0%64)/8)
    const int px  = (s & 1) * 4;        // col group   (d0%8)
    const int aoff = c * HW + py * IMG_W + px;   // compile-time immediate

    v2f a = __builtin_nontemporal_load((const v2f*)(abase + aoff)); // streamed once
    v2f bb = *(const v2f*)(bbase + s * 4);                          // cache-resident

    acc = __builtin_amdgcn_wmma_f32_16x16x4_f32(
        /*neg_a=*/false, a, /*neg_b=*/false, bb,
        /*c_mod=*/(short)0, acc, /*reuse_a=*/false, /*reuse_b=*/false);
  }

  // D layout: VGPR r, lane L -> M = r + 8*khi, N = nlo.
  const int rowbase = n * PATCHES_PER_IMG + pr * PATCH_PER_ROW + pc0 + khi * 8;
  float* obase = out + (size_t)rowbase * DOUT + nlo;
#pragma unroll
  for (int r = 0; r < 8; ++r) {
    __builtin_nontemporal_store(acc[r], obase + r * DOUT);
  }
}

extern "C" void kernel_launch(void* const* d_in, const int* in_sizes, int n_in,
                              void* d_out, int out_size, void* d_ws, size_t ws_size,
                              hipStream_t stream) {
  const float* img  = (const float*)d_in[0];  // 64*3*512*512 f32
  const float* w    = (const float*)d_in[1];  // 16*192 f32
  const float* bias = (const float*)d_in[2];  // 16 f32
  float* out = (float*)d_out;                 // 64*4096*16 f32

  const int total_tiles = 64 * TILES_PER_IMG;         // 16384
  const int blocks = total_tiles / WAVES_PER_BLOCK;   // 2048
  vit_patch_embed_wmma<<<blocks, 256, 0, stream>>>(img, w, bias, out);
}